// GGCM_59279138620019
// MI455X (gfx1250) — compile-verified
//
#include <hip/hip_runtime.h>
#include <hip/hip_bf16.h>
#include <math.h>

#define NODES   100000
#define NEDGE   600000
#define FDIM    128
#define NLAYER  16
#define ALPHA_C 0.15
#define DECL_P  0.9
#define DECL_N  0.5
#define CHUNK   256

typedef __attribute__((address_space(1))) int gint_t;   // global
typedef __attribute__((address_space(3))) int lint_t;   // LDS

// ---------------- CDNA5 async global->LDS helpers ----------------
__device__ __forceinline__ void async_copy_b32(const void* g, void* l) {
#if __has_builtin(__builtin_amdgcn_global_load_async_to_lds_b32)
  __builtin_amdgcn_global_load_async_to_lds_b32(
      (gint_t*)g, (lint_t*)l, /*offset=*/0, /*cpol=*/0);
#else
  unsigned loff = (unsigned)(unsigned long long)(lint_t*)l;
  asm volatile("global_load_async_to_lds_b32 %0, %1, off"
               :: "v"(loff), "v"(g) : "memory");
#endif
}

template <int N>
__device__ __forceinline__ void wait_asynccnt() {
#if __has_builtin(__builtin_amdgcn_s_wait_asynccnt)
  __builtin_amdgcn_s_wait_asynccnt(N);
#else
  asm volatile("s_wait_asynccnt %0" :: "i"(N) : "memory");
#endif
}

// ---------------- setup kernels ----------------
__global__ void k_degree(const int* __restrict__ row, float* __restrict__ deg, int ne) {
  int e = blockIdx.x * blockDim.x + threadIdx.x;
  if (e < ne) unsafeAtomicAdd(&deg[row[e]], 1.0f);
}

__global__ void k_dis(const float* __restrict__ deg, float* __restrict__ dis, int n) {
  int i = blockIdx.x * blockDim.x + threadIdx.x;
  if (i < n) { float d = deg[i]; dis[i] = d > 0.0f ? rsqrtf(d) : 0.0f; }
}

__global__ void k_vals(const int* __restrict__ row, const int* __restrict__ col,
                       const float* __restrict__ dis, float* __restrict__ vals, int ne) {
  int e = blockIdx.x * blockDim.x + threadIdx.x;
  if (e < ne) vals[e] = dis[row[e]] * dis[col[e]];
}

// ---------------- per-layer kernels ----------------
// y = (1-b)*x ; acc = (first ? 0 : acc) + accCoef * x   (fused streaming pass)
__global__ void k_init(const float* __restrict__ x, float* __restrict__ y,
                       float* __restrict__ acc, float oneMinusB, float accCoef,
                       int first, int n4) {
  int i = blockIdx.x * blockDim.x + threadIdx.x;
  if (i >= n4) return;
  float4 xv = ((const float4*)x)[i];
  float4 yv = make_float4(oneMinusB * xv.x, oneMinusB * xv.y,
                          oneMinusB * xv.z, oneMinusB * xv.w);
  ((float4*)y)[i] = yv;
  float4 a = first ? make_float4(0.f, 0.f, 0.f, 0.f) : ((const float4*)acc)[i];
  a.x += accCoef * xv.x; a.y += accCoef * xv.y;
  a.z += accCoef * xv.z; a.w += accCoef * xv.w;
  ((float4*)acc)[i] = a;
}

// acc[nr] -= negCoef * x[nc]  over 120 mirrored negative edges (one wave per edge)
__global__ void k_neg(const int* __restrict__ negIdx, const float* __restrict__ x,
                      float* __restrict__ acc, float negCoef) {
  int t = blockIdx.x * blockDim.x + threadIdx.x;
  int em = t >> 5, lane = t & 31;
  if (em >= 120) return;
  int j = em % 60, dir = em / 60;
  int dst = dir ? negIdx[60 + j] : negIdx[j];        // nr = [idx0, idx1]
  int src = dir ? negIdx[j]      : negIdx[60 + j];   // nc = [idx1, idx0]
  const float4 xv = ((const float4*)x)[(size_t)src * (FDIM / 4) + lane];
  float* ap = acc + (size_t)dst * FDIM + lane * 4;
  unsafeAtomicAdd(ap + 0, -negCoef * xv.x);
  unsafeAtomicAdd(ap + 1, -negCoef * xv.y);
  unsafeAtomicAdd(ap + 2, -negCoef * xv.z);
  unsafeAtomicAdd(ap + 3, -negCoef * xv.w);
}

// y[row[e]] += b*vals[e]*x[col[e]] — edge stream staged to LDS via async DMA,
// double-buffered; one wave covers one 512B feature row (32 lanes x float4).
__global__ __launch_bounds__(256, 2) void k_scatter(
    const int* __restrict__ row, const int* __restrict__ col,
    const float* __restrict__ vals, const float* __restrict__ x,
    float* __restrict__ y, float beta, int ne) {
  __shared__ int   s_row[2][CHUNK];
  __shared__ int   s_col[2][CHUNK];
  __shared__ float s_val[2][CHUNK];

  const int tid = threadIdx.x;
  const int wv = tid >> 5, lane = tid & 31;
  const int stride = (int)gridDim.x * CHUNK;
  int base = (int)blockIdx.x * CHUNK;
  if (base >= ne) return;

  {  // prime pipeline: stage first chunk
    int idx = base + tid; if (idx >= ne) idx = ne - 1;
    async_copy_b32(row  + idx, &s_row[0][tid]);
    async_copy_b32(col  + idx, &s_col[0][tid]);
    async_copy_b32(vals + idx, &s_val[0][tid]);
  }

  int ibuf = 0;
  for (; base < ne; base += stride, ibuf ^= 1) {
    const int nbase = base + stride;
    if (nbase < ne) {  // stage next chunk, then wait only for current's 3 loads
      int idx = nbase + tid; if (idx >= ne) idx = ne - 1;
      async_copy_b32(row  + idx, &s_row[ibuf ^ 1][tid]);
      async_copy_b32(col  + idx, &s_col[ibuf ^ 1][tid]);
      async_copy_b32(vals + idx, &s_val[ibuf ^ 1][tid]);
      wait_asynccnt<3>();
    } else {
      wait_asynccnt<0>();
    }
    __syncthreads();

    const int cnt = min(CHUNK, ne - base);
    for (int j = wv; j < cnt; j += CHUNK / 32) {
      const int   r = s_row[ibuf][j];           // uniform per wave -> LDS broadcast
      const int   c = s_col[ibuf][j];
      const float v = s_val[ibuf][j] * beta;
      const float4 xv = ((const float4*)x)[(size_t)c * (FDIM / 4) + lane];
      float* yp = y + (size_t)r * FDIM + lane * 4;
      unsafeAtomicAdd(yp + 0, v * xv.x);
      unsafeAtomicAdd(yp + 1, v * xv.y);
      unsafeAtomicAdd(yp + 2, v * xv.z);
      unsafeAtomicAdd(yp + 3, v * xv.w);
    }
    __syncthreads();  // buffer ibuf safe to restage next iteration
  }
}

// out = ALPHA*feat + (1-ALPHA)/16 * (acc + 0.5*x_final)
__global__ void k_out(const float* __restrict__ feat, const float* __restrict__ acc,
                      const float* __restrict__ xf, float* __restrict__ out,
                      float aC, float bC, int n4) {
  int i = blockIdx.x * blockDim.x + threadIdx.x;
  if (i >= n4) return;
  float4 f = ((const float4*)feat)[i];
  float4 a = ((const float4*)acc)[i];
  float4 xv = ((const float4*)xf)[i];
  float4 o;
  o.x = aC * f.x + bC * (a.x + 0.5f * xv.x);
  o.y = aC * f.y + bC * (a.y + 0.5f * xv.y);
  o.z = aC * f.z + bC * (a.z + 0.5f * xv.z);
  o.w = aC * f.w + bC * (a.w + 0.5f * xv.w);
  ((float4*)out)[i] = o;
}

// ---------------- host orchestration ----------------
extern "C" void kernel_launch(void* const* d_in, const int* in_sizes, int n_in,
                              void* d_out, int out_size, void* d_ws, size_t ws_size,
                              hipStream_t stream) {
  const float* feat   = (const float*)d_in[0];
  const int*   eidx   = (const int*)d_in[1];
  const int*   negidx = (const int*)d_in[2];
  const int* row = eidx;
  const int* col = eidx + NEDGE;

  float* W    = (float*)d_ws;
  float* bufA = W;                                 // N*D
  float* bufB = bufA + (size_t)NODES * FDIM;       // N*D
  float* acc  = bufB + (size_t)NODES * FDIM;       // N*D
  float* vals = acc  + (size_t)NODES * FDIM;       // E
  float* deg  = vals + NEDGE;                      // N
  float* dis  = deg  + NODES;                      // N

  (void)hipMemsetAsync(deg, 0, NODES * sizeof(float), stream);

  dim3 blk(256);
  k_degree<<<dim3((NEDGE + 255) / 256), blk, 0, stream>>>(row, deg, NEDGE);
  k_dis   <<<dim3((NODES + 255) / 256), blk, 0, stream>>>(deg, dis, NODES);
  k_vals  <<<dim3((NEDGE + 255) / 256), blk, 0, stream>>>(row, col, dis, vals, NEDGE);

  const double c  = (double)(2 * 60) / (double)NODES + 2.0;   // 2.0012
  const int    n4 = NODES * FDIM / 4;

  const float* x = feat;
  float* ybufs[2] = {bufA, bufB};
  double b = 1.0, bn = 1.0;

  for (int l = 0; l < NLAYER; ++l) {
    float* y = ybufs[l & 1];
    const double cinv    = (1.0 - bn) + 2.0 * bn / c;
    const float  accCoef = (float)((l == 0) ? 0.5 * cinv : 0.5 * (1.0 + cinv));
    const float  negCoef = (float)(0.5 * bn / c);

    k_init<<<dim3((n4 + 255) / 256), blk, 0, stream>>>(
        x, y, acc, (float)(1.0 - b), accCoef, (l == 0) ? 1 : 0, n4);
    k_neg<<<dim3((120 * 32 + 255) / 256), blk, 0, stream>>>(
        negidx + l * 120, x, acc, negCoef);
    k_scatter<<<dim3(2048), blk, 0, stream>>>(
        row, col, vals, x, y, (float)b, NEDGE);

    x = y;
    b *= DECL_P;
    bn *= DECL_N;
  }

  k_out<<<dim3((n4 + 255) / 256), blk, 0, stream>>>(
      feat, acc, x, (float*)d_out,
      (float)ALPHA_C, (float)((1.0 - ALPHA_C) / (double)NLAYER), n4);
}